// SparseMoeBlock_31928786879172
// MI455X (gfx1250) — compile-verified
//
#include <hip/hip_runtime.h>
#include <cstdint>
#include <cstddef>

// ---------------------------------------------------------------------------
// MoE block (8 experts, top-2, H=2048, I=1408, T=8192), all fp32.
// Memory-bound (~410MB compulsory @ 23.3TB/s); fp32 WMMA (16x16x4) keeps
// bit-compatible precision with the fp32 reference while using the matrix
// pipe. M=tokens, N=features, A=activations, B=weights; D stores coalesced.
// Double-buffered LDS staging (global loads issued before the WMMA block
// each iteration), b128-vectorized fills (4x fewer issue slots), padded
// conflict-free LDS pitch, v_rcp_f32 SiLU epilogue.
// ---------------------------------------------------------------------------

#define T_TOKENS 8192
#define HIDDEN   2048
#define INTER    1408
#define NEXP     8

#define KC   32   // K chunk staged through LDS
#define LDP  36   // padded LDS row stride (floats): 144B -> b128-aligned,
                  // and 36*r mod 64 banks all-distinct for r=0..15 (no conflicts)

typedef float v2f __attribute__((ext_vector_type(2)));
typedef float v4f __attribute__((ext_vector_type(4)));
typedef float v8f __attribute__((ext_vector_type(8)));

// ---------------------------------------------------------------------------
// Router: logits -> softmax -> top2 -> renormalize -> dense combine[T][8].
// One wave per token; 8 accumulators per lane; shuffle reduction.
// ---------------------------------------------------------------------------
__global__ __launch_bounds__(256) void moe_router(
        const float* __restrict__ x,        // [T][H]
        const float* __restrict__ rw,       // [E][H]
        float* __restrict__ combine) {      // [T][E]
    const int lane = threadIdx.x & 31;
    const int wave = threadIdx.x >> 5;
    const int t = blockIdx.x * 8 + wave;

    float acc[NEXP];
#pragma unroll
    for (int e = 0; e < NEXP; ++e) acc[e] = 0.0f;

    const float* xrow = x + (size_t)t * HIDDEN;
    for (int h = lane; h < HIDDEN; h += 32) {
        const float xv = xrow[h];
#pragma unroll
        for (int e = 0; e < NEXP; ++e) acc[e] += xv * rw[e * HIDDEN + h];
    }
#pragma unroll
    for (int e = 0; e < NEXP; ++e) {
#pragma unroll
        for (int off = 16; off >= 1; off >>= 1)
            acc[e] += __shfl_xor(acc[e], off, 32);
    }

    if (lane == 0) {
        // top-2 by logit (== top-2 by prob); ties -> lowest index (jax top_k)
        int i1 = 0;
#pragma unroll
        for (int e = 1; e < NEXP; ++e) if (acc[e] > acc[i1]) i1 = e;
        int i2 = (i1 == 0) ? 1 : 0;
#pragma unroll
        for (int e = 0; e < NEXP; ++e)
            if (e != i1 && acc[e] > acc[i2]) i2 = e;
        // p1/(p1+p2) = 1/(1+exp(l2-l1)) : stable, skips full softmax
        const float w1 = 1.0f / (1.0f + __expf(acc[i2] - acc[i1]));
        const float w2 = 1.0f - w1;
        float* crow = combine + (size_t)t * NEXP;
#pragma unroll
        for (int e = 0; e < NEXP; ++e) crow[e] = 0.0f;
        crow[i1] = w1;
        crow[i2] = w2;
    }
}

// ---------------------------------------------------------------------------
// gate_up + SiLU*up for one expert: hbuf[t][i] = silu(x.Wg^T) * (x.Wu^T)
// Block: 256 thr = 8 waves; tile = 16 tokens x 128 inter cols (16x16/wave).
// fp32 A/B lane layout: element k-index = kbase + r + 2*(lane>=16).
// ---------------------------------------------------------------------------
__global__ __launch_bounds__(256) void moe_gateup(
        const float* __restrict__ x,        // [T][H]
        const float* __restrict__ gup,      // [2I][H] (expert pre-offset)
        float* __restrict__ hbuf) {         // [T][I]
    __shared__ float Xs[2][16][LDP];
    __shared__ float Gs[2][128][LDP];
    __shared__ float Us[2][128][LDP];

    const int tid   = threadIdx.x;
    const int lane  = tid & 31;
    const int wave  = tid >> 5;
    const int lg    = lane >> 4;   // half-wave group
    const int lm    = lane & 15;
    const int t0    = blockIdx.x * 16;
    const int n0blk = blockIdx.y * 128;
    const int n0    = n0blk + wave * 16;

    const int q4    = (tid & 7) * 4;   // float4 column within K chunk
    const int rowp  = tid >> 3;        // 0..31 row phase (uniform per wave /8)

    v4f xr = {};
    v4f gr[4], ur[4];

    // prologue: stage chunk kk=0 (b128 loads) and commit to buffer 0
    if (rowp < 16)
        xr = *(const v4f*)&x[(size_t)(t0 + rowp) * HIDDEN + q4];
#pragma unroll
    for (int j = 0; j < 4; ++j) {
        const int rr = rowp + 32 * j;
        gr[j] = *(const v4f*)&gup[(size_t)(n0blk + rr) * HIDDEN + q4];
        ur[j] = *(const v4f*)&gup[(size_t)(INTER + n0blk + rr) * HIDDEN + q4];
    }
    if (rowp < 16) *(v4f*)&Xs[0][rowp][q4] = xr;
#pragma unroll
    for (int j = 0; j < 4; ++j) {
        const int rr = rowp + 32 * j;
        *(v4f*)&Gs[0][rr][q4] = gr[j];
        *(v4f*)&Us[0][rr][q4] = ur[j];
    }

    v8f dg = {};
    v8f du = {};
    int cur = 0;

    for (int kk = 0; kk < HIDDEN; kk += KC) {
        __syncthreads();                       // buf[cur] ready for all waves
        const int kn = kk + KC;
        if (kn < HIDDEN) {                     // issue next chunk's loads now
            if (rowp < 16)
                xr = *(const v4f*)&x[(size_t)(t0 + rowp) * HIDDEN + kn + q4];
#pragma unroll
            for (int j = 0; j < 4; ++j) {
                const int rr = rowp + 32 * j;
                gr[j] = *(const v4f*)&gup[(size_t)(n0blk + rr) * HIDDEN + kn + q4];
                ur[j] = *(const v4f*)&gup[(size_t)(INTER + n0blk + rr) * HIDDEN + kn + q4];
            }
        }
#pragma unroll
        for (int k4 = 0; k4 < KC; k4 += 4) {   // WMMA on current buffer
            const int ks = k4 + 2 * lg;
            const v2f a  = *(const v2f*)&Xs[cur][lm][ks];
            const v2f bg = *(const v2f*)&Gs[cur][wave * 16 + lm][ks];
            const v2f bu = *(const v2f*)&Us[cur][wave * 16 + lm][ks];
            dg = __builtin_amdgcn_wmma_f32_16x16x4_f32(
                     false, a, false, bg, (short)0, dg, false, false);
            du = __builtin_amdgcn_wmma_f32_16x16x4_f32(
                     false, a, false, bu, (short)0, du, false, false);
        }
        if (kn < HIDDEN) {                     // commit staged chunk (b128)
            const int nxt = cur ^ 1;
            if (rowp < 16) *(v4f*)&Xs[nxt][rowp][q4] = xr;
#pragma unroll
            for (int j = 0; j < 4; ++j) {
                const int rr = rowp + 32 * j;
                *(v4f*)&Gs[nxt][rr][q4] = gr[j];
                *(v4f*)&Us[nxt][rr][q4] = ur[j];
            }
        }
        cur ^= 1;
    }

    // D[r]: token m = r + 8*lg, inter col n = lm  -> coalesced over lanes
#pragma unroll
    for (int r = 0; r < 8; ++r) {
        const float g = dg[r];
        const float u = du[r];
        const float sig = __builtin_amdgcn_rcpf(1.0f + __expf(-g));
        hbuf[(size_t)(t0 + r + 8 * lg) * INTER + n0 + lm] = g * sig * u;
    }
}

// ---------------------------------------------------------------------------
// down proj + weighted combine: out[t][h] (+)= combine[t][e] * (h . Wd[h])
// Same double-buffered tiling; expert 0 writes, others accumulate.
// ---------------------------------------------------------------------------
__global__ __launch_bounds__(256) void moe_down(
        const float* __restrict__ hbuf,     // [T][I]
        const float* __restrict__ wd,       // [H][I] (expert pre-offset)
        const float* __restrict__ combine,  // [T][E]
        float* __restrict__ out,            // [T][H]
        int expert, int accumulate) {
    __shared__ float Hs[2][16][LDP];
    __shared__ float Ws[2][128][LDP];
    __shared__ float cw[16];

    const int tid   = threadIdx.x;
    const int lane  = tid & 31;
    const int wave  = tid >> 5;
    const int lg    = lane >> 4;
    const int lm    = lane & 15;
    const int t0    = blockIdx.x * 16;
    const int n0blk = blockIdx.y * 128;
    const int n0    = n0blk + wave * 16;

    if (tid < 16)
        cw[tid] = combine[(size_t)(t0 + tid) * NEXP + expert];

    const int q4   = (tid & 7) * 4;
    const int rowp = tid >> 3;

    v4f hr = {};
    v4f wr[4];

    // prologue: stage chunk kk=0 into buffer 0
    if (rowp < 16)
        hr = *(const v4f*)&hbuf[(size_t)(t0 + rowp) * INTER + q4];
#pragma unroll
    for (int j = 0; j < 4; ++j)
        wr[j] = *(const v4f*)&wd[(size_t)(n0blk + rowp + 32 * j) * INTER + q4];
    if (rowp < 16) *(v4f*)&Hs[0][rowp][q4] = hr;
#pragma unroll
    for (int j = 0; j < 4; ++j)
        *(v4f*)&Ws[0][rowp + 32 * j][q4] = wr[j];

    v8f d = {};
    int cur = 0;

    for (int kk = 0; kk < INTER; kk += KC) {
        __syncthreads();   // also fences the cw[] write on first iteration
        const int kn = kk + KC;
        if (kn < INTER) {
            if (rowp < 16)
                hr = *(const v4f*)&hbuf[(size_t)(t0 + rowp) * INTER + kn + q4];
#pragma unroll
            for (int j = 0; j < 4; ++j)
                wr[j] = *(const v4f*)&wd[(size_t)(n0blk + rowp + 32 * j) * INTER + kn + q4];
        }
#pragma unroll
        for (int k4 = 0; k4 < KC; k4 += 4) {
            const int ks = k4 + 2 * lg;
            const v2f a = *(const v2f*)&Hs[cur][lm][ks];
            const v2f b = *(const v2f*)&Ws[cur][wave * 16 + lm][ks];
            d = __builtin_amdgcn_wmma_f32_16x16x4_f32(
                    false, a, false, b, (short)0, d, false, false);
        }
        if (kn < INTER) {
            const int nxt = cur ^ 1;
            if (rowp < 16) *(v4f*)&Hs[nxt][rowp][q4] = hr;
#pragma unroll
            for (int j = 0; j < 4; ++j)
                *(v4f*)&Ws[nxt][rowp + 32 * j][q4] = wr[j];
        }
        cur ^= 1;
    }

#pragma unroll
    for (int r = 0; r < 8; ++r) {
        const int m = r + 8 * lg;
        const float w = cw[m];
        const size_t oidx = (size_t)(t0 + m) * HIDDEN + n0 + lm;
        const float v = w * d[r];
        if (accumulate) out[oidx] += v;
        else            out[oidx] = v;
    }
}

// ---------------------------------------------------------------------------
// Host launcher. Workspace: combine[T][8] (256KB) + hbuf[T][I] (46.1MB).
// Experts processed sequentially so hbuf is reused (stream-ordered); hbuf
// fits in the 192MB L2 between the gateup and down kernels of each expert.
// ---------------------------------------------------------------------------
extern "C" void kernel_launch(void* const* d_in, const int* in_sizes, int n_in,
                              void* d_out, int out_size, void* d_ws, size_t ws_size,
                              hipStream_t stream) {
    (void)in_sizes; (void)n_in; (void)out_size; (void)ws_size;

    const float* x   = (const float*)d_in[0];   // hidden_states [T][H]
    const float* rw  = (const float*)d_in[1];   // router_weight [E][H]
    const float* gup = (const float*)d_in[2];   // gate_up_proj  [E][2I][H]
    const float* dwn = (const float*)d_in[3];   // down_proj     [E][H][I]
    float* out = (float*)d_out;                 // [T][H]

    float* combine = (float*)d_ws;                         // T*E floats
    float* hbuf    = combine + (size_t)T_TOKENS * NEXP;    // T*I floats

    moe_router<<<T_TOKENS / 8, 256, 0, stream>>>(x, rw, combine);

    for (int e = 0; e < NEXP; ++e) {
        const float* gup_e = gup + (size_t)e * 2 * INTER * HIDDEN;
        const float* wd_e  = dwn + (size_t)e * HIDDEN * INTER;

        moe_gateup<<<dim3(T_TOKENS / 16, INTER / 128), 256, 0, stream>>>(
            x, gup_e, hbuf);

        moe_down<<<dim3(T_TOKENS / 16, HIDDEN / 128), 256, 0, stream>>>(
            hbuf, wd_e, combine, out, e, e != 0);
    }
}